// DropLearner_34368328303097
// MI455X (gfx1250) — compile-verified
//
#include <hip/hip_runtime.h>
#include <hip/hip_bf16.h>

typedef __attribute__((ext_vector_type(2))) float v2f;
typedef __attribute__((ext_vector_type(8))) float v8f;

#define TEMP_DE   0.5f
#define BIASC     0.0001f
#define HDIM      128
#define NBINS     4096

// monotone float -> uint key (larger float => larger uint)
__device__ __forceinline__ unsigned ordkey(float f) {
    unsigned u = __float_as_uint(f);
    return (u & 0x80000000u) ? ~u : (u | 0x80000000u);
}

// ---------------------------------------------------------------------------
// Kernel 1: per-edge dot product via V_WMMA_F32_16X16X4_F32 (diagonal trick)
// One wave32 handles 16 edges. A = x[row[e0..e15]] (16xK), B = x[col[..]] (Kx16).
// Accumulate C over K=128 in 32 steps of K=4; dot[i] = C[i][i].
// A-fragment layout (ISA 7.12.2): lane L, VGPR v -> [M=L%16][K = v + 2*(L>>4)]
// B-fragment mirrors it, so each lane loads one contiguous float2 per source.
// ---------------------------------------------------------------------------
__global__ void k_dot_logit(const float* __restrict__ x,
                            const long long* __restrict__ row,
                            const long long* __restrict__ col,
                            const float* __restrict__ noise,
                            float* __restrict__ s_ws, int E) {
    const int lane  = threadIdx.x & 31;
    const int wave  = blockIdx.x * (blockDim.x >> 5) + (threadIdx.x >> 5);
    const int ebase = wave * 16;
    if (ebase >= E) return;                 // uniform per wave: EXEC stays all-1s

    const int el = lane & 15;               // which of the 16 edges this lane serves
    const int e  = ebase + el;
    const int ec = (e < E) ? e : (E - 1);   // clamp (no divergence around WMMA)

    const long long r = row[ec];
    const long long c = col[ec];
    const float* xr = x + r * (long long)HDIM;
    const float* xc = x + c * (long long)HDIM;
    const int koff = (lane >> 4) << 1;      // 0 for lanes 0-15, 2 for lanes 16-31

    __builtin_prefetch(xr, 0, 0);
    __builtin_prefetch(xc, 0, 0);
    __builtin_prefetch(xr + 64, 0, 0);
    __builtin_prefetch(xc + 64, 0, 0);

    v8f acc = {};
#pragma unroll
    for (int k0 = 0; k0 < HDIM; k0 += 4) {
        v2f a = *(const v2f*)(xr + k0 + koff);   // 8B gather, 8B-aligned
        v2f b = *(const v2f*)(xc + k0 + koff);
        acc = __builtin_amdgcn_wmma_f32_16x16x4_f32(
            /*neg_a=*/false, a, /*neg_b=*/false, b,
            /*c_mod=*/(short)0, acc, /*reuse_a=*/false, /*reuse_b=*/false);
    }

    // Diagonal C[i][i]: lanes 0-7 (edges 0-7) at VGPR=lane, lanes 24-31
    // (edges 8-15) at VGPR=lane-24.
    float wgt = 0.f;
#pragma unroll
    for (int p = 0; p < 8; ++p) {
        bool d = (lane < 16) ? (lane == p) : (lane == p + 24);
        if (d) wgt = acc[p];
    }

    const bool active = (lane < 8) || (lane >= 24);
    if (active && e < E) {
        float nz  = noise[e];
        float eps = (2.f * BIASC - 1.f) * nz + (1.f - BIASC);
        float ln  = __logf(eps) - __logf(1.f - eps);
        s_ws[e]   = (ln + wgt) / TEMP_DE;    // pre-sigmoid logit (monotone in gate)
    }
}

// ---------------------------------------------------------------------------
// Workspace zeroing
// ---------------------------------------------------------------------------
__global__ void k_zero(unsigned* __restrict__ p, int n) {
    int i = blockIdx.x * blockDim.x + threadIdx.x;
    if (i < n) p[i] = 0u;
}

// ---------------------------------------------------------------------------
// Radix-select level 1: histogram of top 12 bits of ordered key
// ---------------------------------------------------------------------------
__global__ void k_hist1(const float* __restrict__ s, unsigned* __restrict__ hist, int E) {
    __shared__ unsigned h[NBINS];
    for (int i = threadIdx.x; i < NBINS; i += blockDim.x) h[i] = 0u;
    __syncthreads();
    for (int e = blockIdx.x * blockDim.x + threadIdx.x; e < E; e += gridDim.x * blockDim.x)
        atomicAdd(&h[ordkey(s[e]) >> 20], 1u);
    __syncthreads();
    for (int i = threadIdx.x; i < NBINS; i += blockDim.x)
        if (h[i]) atomicAdd(&hist[i], h[i]);
}

__global__ void k_scan1(const unsigned* __restrict__ hist, unsigned* __restrict__ ctrl, int k) {
    if (threadIdx.x == 0 && blockIdx.x == 0) {
        unsigned cum = 0; unsigned b = 0; unsigned rank = (unsigned)k;
        for (int i = NBINS - 1; i >= 0; --i) {
            unsigned c = hist[i];
            if (cum + c >= (unsigned)k) { b = (unsigned)i; rank = (unsigned)k - cum; break; }
            cum += c;
        }
        ctrl[0] = b;      // bin of k-th largest (top 12 bits)
        ctrl[1] = rank;   // 1-based rank inside that bin
    }
}

// ---------------------------------------------------------------------------
// Radix-select level 2: histogram next 12 bits, elements inside bin ctrl[0]
// ---------------------------------------------------------------------------
__global__ void k_hist2(const float* __restrict__ s, const unsigned* __restrict__ ctrl,
                        unsigned* __restrict__ hist2, int E) {
    __shared__ unsigned h[NBINS];
    for (int i = threadIdx.x; i < NBINS; i += blockDim.x) h[i] = 0u;
    __syncthreads();
    const unsigned b1 = ctrl[0];
    for (int e = blockIdx.x * blockDim.x + threadIdx.x; e < E; e += gridDim.x * blockDim.x) {
        unsigned u = ordkey(s[e]);
        if ((u >> 20) == b1) atomicAdd(&h[(u >> 8) & 0xFFFu], 1u);
    }
    __syncthreads();
    for (int i = threadIdx.x; i < NBINS; i += blockDim.x)
        if (h[i]) atomicAdd(&hist2[i], h[i]);
}

__global__ void k_scan2(const unsigned* __restrict__ hist2, unsigned* __restrict__ ctrl) {
    if (threadIdx.x == 0 && blockIdx.x == 0) {
        const unsigned rank = ctrl[1];
        const unsigned b1   = ctrl[0];
        unsigned cum = 0; unsigned b2 = 0;
        for (int i = NBINS - 1; i >= 0; --i) {
            unsigned c = hist2[i];
            if (cum + c >= rank) { b2 = (unsigned)i; break; }
            cum += c;
        }
        ctrl[2] = (b1 << 20) | (b2 << 8);   // 24-bit keep threshold: keep if key >= T
    }
}

// ---------------------------------------------------------------------------
// Gate + rowsum (segment-sum over `row` via float atomics)
// ---------------------------------------------------------------------------
__global__ void k_gate_rowsum(const float* __restrict__ s_ws,
                              const long long* __restrict__ row,
                              const unsigned* __restrict__ ctrl,
                              float* __restrict__ gate_out,
                              float* __restrict__ rowsum, int E) {
    int e = blockIdx.x * blockDim.x + threadIdx.x;
    if (e >= E) return;
    float s = s_ws[e];
    const unsigned T = ctrl[2];
    float g = (ordkey(s) >= T) ? (1.f / (1.f + __expf(-s))) : 0.f;
    gate_out[e] = g;
    if (g != 0.f) atomicAdd(&rowsum[row[e]], g);
}

__global__ void k_dinv(const float* __restrict__ rowsum, float* __restrict__ dinv, int N) {
    int n = blockIdx.x * blockDim.x + threadIdx.x;
    if (n >= N) return;
    float rs = rowsum[n];
    dinv[n] = (rs > 0.f) ? fminf(rsqrtf(rs), 10.f) : 10.f;
}

__global__ void k_final(const long long* __restrict__ row,
                        const long long* __restrict__ col,
                        const float* __restrict__ dinv,
                        float* __restrict__ out, int E) {
    int e = blockIdx.x * blockDim.x + threadIdx.x;
    if (e >= E) return;
    float g = out[e];
    out[e] = g * dinv[row[e]] * dinv[col[e]];
}

// ---------------------------------------------------------------------------
extern "C" void kernel_launch(void* const* d_in, const int* in_sizes, int n_in,
                              void* d_out, int out_size, void* d_ws, size_t ws_size,
                              hipStream_t stream) {
    const float*     x     = (const float*)d_in[0];
    const long long* ei    = (const long long*)d_in[1];   // int64 [2, E]
    const float*     noise = (const float*)d_in[2];
    const int N = in_sizes[0] / HDIM;
    const int E = in_sizes[2];
    const long long* row = ei;
    const long long* col = ei + E;
    float* out = (float*)d_out;

    // workspace layout (floats/u32 words)
    float*    s_ws   = (float*)d_ws;          // [E]
    float*    rowsum = s_ws + E;              // [N]
    float*    dinv   = rowsum + N;            // [N]
    unsigned* hist1  = (unsigned*)(dinv + N); // [4096]
    unsigned* hist2  = hist1 + NBINS;         // [4096]
    unsigned* ctrl   = hist2 + NBINS;         // [8]

    const int k = (int)((double)E * 0.8);     // static top-k count

    // zero rowsum..ctrl (dinv in between is overwritten anyway)
    const int zw = 2 * N + 2 * NBINS + 8;
    k_zero<<<(zw + 255) / 256, 256, 0, stream>>>((unsigned*)rowsum, zw);

    // 1) WMMA dot products + gumbel logits
    {
        const int waves  = (E + 15) / 16;
        const int blocks = (waves + 7) / 8;   // 8 waves (256 threads) per block
        k_dot_logit<<<blocks, 256, 0, stream>>>(x, row, col, noise, s_ws, E);
    }

    // 2) two-level radix select for the top-k threshold
    k_hist1<<<512, 256, 0, stream>>>(s_ws, hist1, E);
    k_scan1<<<1, 32, 0, stream>>>(hist1, ctrl, k);
    k_hist2<<<512, 256, 0, stream>>>(s_ws, ctrl, hist2, E);
    k_scan2<<<1, 32, 0, stream>>>(hist2, ctrl);

    // 3) gate + degree accumulation, 4) D^-1/2, 5) normalize edge values
    k_gate_rowsum<<<(E + 255) / 256, 256, 0, stream>>>(s_ws, row, ctrl, out, rowsum, E);
    k_dinv<<<(N + 255) / 256, 256, 0, stream>>>(rowsum, dinv, N);
    k_final<<<(E + 255) / 256, 256, 0, stream>>>(row, col, dinv, out, E);
}